// Components_19198503813817
// MI455X (gfx1250) — compile-verified
//
#include <hip/hip_runtime.h>
#include <hip/hip_bf16.h>
#include <stdint.h>

#define TILE 32
#define SCAN_TPB 256
#define SCAN_EPT 16
#define SCAN_EPB (SCAN_TPB * SCAN_EPT) /* 4096 */
#define BG_SENTINEL 0x7FFFFFFF         /* background marker in parent array */

// ---------------------------------------------------------------------------
// Lock-free union-find merge (union by min index). Invariant: P[x] <= x for
// foreground entries, so the final root of every component is its minimum
// flat index -> matches the reference's min-label semantics deterministically.
// ---------------------------------------------------------------------------
__device__ __forceinline__ void unite(int* __restrict__ P, int a, int b) {
  while (true) {
    int pa = P[a];
    while (pa != a) { a = pa; pa = P[a]; }
    int pb = P[b];
    while (pb != b) { b = pb; pb = P[b]; }
    if (a == b) return;
    const int lo = min(a, b);
    const int hi = max(a, b);
    const int old = atomicMin(&P[hi], lo);
    if (old == hi) return;   // successfully linked hi -> lo
    a = lo; b = old;         // hi already pointed to old; merge lo with old
  }
}

// ---------------------------------------------------------------------------
// Kernel 1: per-tile CCL in LDS. Input tile staged via the CDNA5 async
// global->LDS DMA path (ASYNCcnt). Writes the parent forest into P:
//   foreground pixel -> global index of its tile-local min root
//   background pixel -> BG_SENTINEL
// ---------------------------------------------------------------------------
__global__ __launch_bounds__(TILE * TILE) void local_ccl_kernel(
    const float* __restrict__ in, int* __restrict__ P, int W, int H) {
  __shared__ __align__(16) float smIn[TILE * TILE];
  __shared__ int lbl[TILE * TILE];
  __shared__ int changed;

  const int lx = threadIdx.x, ly = threadIdx.y;
  const int t = ly * TILE + lx;
  const int b = blockIdx.z;
  const int y = blockIdx.y * TILE + ly;
  const int x = blockIdx.x * TILE + lx;
  const int g = (b * H + y) * W + x;

  // ---- gfx1250 async global->LDS staging: 256 lanes x 16B cover the tile ---
  if (t < 256) {
    const int row = t >> 3;            // 0..31
    const int col = (t & 7) << 2;      // 0,4,...,28 (float4 granularity)
    const int gy = blockIdx.y * TILE + row;
    const int gx = blockIdx.x * TILE + col;
    const uint32_t goff = (uint32_t)(((b * H + gy) * W + gx) * 4);
    const uint32_t loff = (uint32_t)(uintptr_t)(&smIn[row * TILE + col]);
    asm volatile("global_load_async_to_lds_b128 %0, %1, %2"
                 :: "v"(loff), "v"(goff), "s"(in) : "memory");
  }
  asm volatile("s_wait_asynccnt 0" ::: "memory");
  __syncthreads();

  const bool m = smIn[t] > 0.0f;
  lbl[t] = m ? t : (TILE * TILE);      // sentinel for background
  __syncthreads();

  for (;;) {
    if (t == 0) changed = 0;
    __syncthreads();                   // A: reset visible
    if (m) {
      const int cur = lbl[t];
      int mn = cur;
      if (lx > 0)        mn = min(mn, lbl[t - 1]);
      if (lx < TILE - 1) mn = min(mn, lbl[t + 1]);
      if (ly > 0)        mn = min(mn, lbl[t - TILE]);
      if (ly < TILE - 1) mn = min(mn, lbl[t + TILE]);
      mn = min(mn, lbl[mn]);           // pointer jump (mn <= cur < 1024)
      if (mn < cur) { lbl[t] = mn; changed = 1; }
    }
    __syncthreads();                   // B: updates visible
    const bool done = (changed == 0);
    __syncthreads();                   // C: all read before next reset
    if (done) break;
  }

  int v = BG_SENTINEL;
  if (m) {
    const int r = lbl[t];              // tile-local min root index
    const int rx = blockIdx.x * TILE + (r & (TILE - 1));
    const int ry = blockIdx.y * TILE + (r >> 5);
    v = (b * H + ry) * W + rx;
  }
  P[g] = v;
}

// ---------------------------------------------------------------------------
// Kernel 2: merge across tile borders (right edge + bottom edge of each tile).
// Only foreground pairs are united; BG_SENTINEL entries are never touched.
// ---------------------------------------------------------------------------
__global__ __launch_bounds__(2 * TILE) void border_merge_kernel(
    const float* __restrict__ in, int* __restrict__ P,
    int W, int H, int tilesX, int tilesY) {
  const int tile = blockIdx.x;
  const int tx  = tile % tilesX;
  const int rem = tile / tilesX;
  const int ty  = rem % tilesY;
  const int b   = rem / tilesY;
  const int t = threadIdx.x;

  int x, y, nx, ny;
  if (t < TILE) {                      // right edge pixel -> neighbor at x+1
    x = tx * TILE + (TILE - 1); y = ty * TILE + t; nx = x + 1; ny = y;
  } else {                             // bottom edge pixel -> neighbor at y+1
    x = tx * TILE + (t - TILE); y = ty * TILE + (TILE - 1); nx = x; ny = y + 1;
  }
  if (nx >= W || ny >= H) return;
  const int g  = (b * H + y)  * W + x;
  const int gn = (b * H + ny) * W + nx;
  if (in[g] > 0.0f && in[gn] > 0.0f) unite(P, g, gn);
}

// ---------------------------------------------------------------------------
// Kernel 3 (fused): full path compression + per-block root counts.
// Roots are fixed after all merges, so concurrent compression is safe.
// flag = (root == i); background (BG_SENTINEL) can never match.
// ---------------------------------------------------------------------------
__global__ __launch_bounds__(SCAN_TPB) void flatten_sums_kernel(
    int* __restrict__ P, int* __restrict__ partials) {
  const int t = threadIdx.x;
  const int base = blockIdx.x * SCAN_EPB + t * SCAN_EPT;
  int r[SCAN_EPT];

  const int4* P4 = (const int4*)(P + base);
#pragma unroll
  for (int q = 0; q < 4; ++q) {
    const int4 p = P4[q];
    r[q * 4 + 0] = p.x; r[q * 4 + 1] = p.y;
    r[q * 4 + 2] = p.z; r[q * 4 + 3] = p.w;
  }

  int s = 0;
#pragma unroll
  for (int e = 0; e < SCAN_EPT; ++e) {
    int root = r[e];
    if (root != BG_SENTINEL) {
      int pr = P[root];
      while (pr != root) { root = pr; pr = P[root]; }
      s += (root == base + e);
      r[e] = root;
    }
  }

  int4* PW = (int4*)(P + base);
#pragma unroll
  for (int q = 0; q < 4; ++q) {
    int4 o;
    o.x = r[q * 4 + 0]; o.y = r[q * 4 + 1];
    o.z = r[q * 4 + 2]; o.w = r[q * 4 + 3];
    PW[q] = o;
  }

  __shared__ int sh[SCAN_TPB];
  sh[t] = s;
  __syncthreads();
#pragma unroll
  for (int off = SCAN_TPB / 2; off > 0; off >>= 1) {
    if (t < off) sh[t] += sh[t + off];
    __syncthreads();
  }
  if (t == 0) partials[blockIdx.x] = sh[0];
}

// ---------------------------------------------------------------------------
// Kernel 4: exclusive scan of per-block root counts (single block, <=4096).
// ---------------------------------------------------------------------------
__global__ __launch_bounds__(SCAN_TPB) void scan_partials_kernel(
    int* __restrict__ partials, int nb) {
  const int t = threadIdx.x;
  const int base = t * SCAN_EPT;
  int vals[SCAN_EPT];
  int s = 0;
#pragma unroll
  for (int k = 0; k < SCAN_EPT; ++k) {
    const int v = (base + k < nb) ? partials[base + k] : 0;
    vals[k] = v; s += v;
  }
  __shared__ int sh[SCAN_TPB];
  sh[t] = s;
  __syncthreads();
#pragma unroll
  for (int off = 1; off < SCAN_TPB; off <<= 1) {
    const int xv = (t >= off) ? sh[t - off] : 0;
    __syncthreads();
    sh[t] += xv;
    __syncthreads();
  }
  int run = sh[t] - s;                 // exclusive prefix for this thread
#pragma unroll
  for (int k = 0; k < SCAN_EPT; ++k) {
    if (base + k < nb) partials[base + k] = run;
    run += vals[k];
  }
}

// ---------------------------------------------------------------------------
// Kernel 5: write -rank into P at root positions only. Each thread writes
// only inside its own 16-element chunk -> no cross-thread hazard.
// ---------------------------------------------------------------------------
__global__ __launch_bounds__(SCAN_TPB) void rank_roots_kernel(
    int* __restrict__ P, const int* __restrict__ partials) {
  const int t = threadIdx.x;
  const int base = blockIdx.x * SCAN_EPB + t * SCAN_EPT;
  const int4* P4 = (const int4*)(P + base);
  int flag[SCAN_EPT];
  int s = 0;
#pragma unroll
  for (int q = 0; q < 4; ++q) {
    const int4 p = P4[q];
    const int i0 = base + q * 4;
    flag[q * 4 + 0] = (p.x == i0 + 0);
    flag[q * 4 + 1] = (p.y == i0 + 1);
    flag[q * 4 + 2] = (p.z == i0 + 2);
    flag[q * 4 + 3] = (p.w == i0 + 3);
    s += flag[q * 4 + 0] + flag[q * 4 + 1] + flag[q * 4 + 2] + flag[q * 4 + 3];
  }
  __shared__ int sh[SCAN_TPB];
  sh[t] = s;
  __syncthreads();
#pragma unroll
  for (int off = 1; off < SCAN_TPB; off <<= 1) {
    const int xv = (t >= off) ? sh[t - off] : 0;
    __syncthreads();
    sh[t] += xv;
    __syncthreads();
  }
  int run = partials[blockIdx.x] + (sh[t] - s);
#pragma unroll
  for (int e = 0; e < SCAN_EPT; ++e) {
    run += flag[e];
    if (flag[e]) P[base + e] = -run;   // inclusive rank >= 1, negated marker
  }
}

// ---------------------------------------------------------------------------
// Kernel 6: final labels straight into d_out.
//   P[i] < 0          -> i is a root, label = -P[i]
//   P[i] == SENTINEL  -> background, label = 0
//   else              -> label = -P[P[i]] (root holds negated rank)
// ---------------------------------------------------------------------------
__global__ void final_label_kernel(const int* __restrict__ P,
                                   int* __restrict__ out, int N) {
  const int i = (blockIdx.x * blockDim.x + threadIdx.x) * 4;
  if (i >= N) return;
  const int4 p = *(const int4*)(P + i);
  int4 o;
  o.x = (p.x < 0) ? -p.x : ((p.x != BG_SENTINEL) ? -P[p.x] : 0);
  o.y = (p.y < 0) ? -p.y : ((p.y != BG_SENTINEL) ? -P[p.y] : 0);
  o.z = (p.z < 0) ? -p.z : ((p.z != BG_SENTINEL) ? -P[p.z] : 0);
  o.w = (p.w < 0) ? -p.w : ((p.w != BG_SENTINEL) ? -P[p.w] : 0);
  *(int4*)(out + i) = o;
}

extern "C" void kernel_launch(void* const* d_in, const int* in_sizes, int n_in,
                              void* d_out, int out_size, void* d_ws, size_t ws_size,
                              hipStream_t stream) {
  (void)n_in; (void)out_size; (void)ws_size;
  const float* in = (const float*)d_in[0];
  const int N = in_sizes[0];
  const int W = 1024, H = 1024;
  const int B = N / (W * H);

  int* P        = (int*)d_ws;                                  // N ints
  int* partials = (int*)((char*)d_ws + (size_t)N * sizeof(int));
  int* out      = (int*)d_out;

  const int tilesX = W / TILE, tilesY = H / TILE;
  const int nb = N / SCAN_EPB;   // 4096 for N = 2^24; fits one 2nd-level block

  dim3 lgrid(tilesX, tilesY, B), lblock(TILE, TILE, 1);
  local_ccl_kernel<<<lgrid, lblock, 0, stream>>>(in, P, W, H);

  border_merge_kernel<<<tilesX * tilesY * B, 2 * TILE, 0, stream>>>(
      in, P, W, H, tilesX, tilesY);

  flatten_sums_kernel<<<nb, SCAN_TPB, 0, stream>>>(P, partials);
  scan_partials_kernel<<<1, SCAN_TPB, 0, stream>>>(partials, nb);
  rank_roots_kernel<<<nb, SCAN_TPB, 0, stream>>>(P, partials);

  final_label_kernel<<<(N / 4 + 255) / 256, 256, 0, stream>>>(P, out, N);
}